// DP_LQR_61469571940706
// MI455X (gfx1250) — compile-verified
//
#include <hip/hip_runtime.h>

typedef __attribute__((ext_vector_type(2))) float v2f;
typedef __attribute__((ext_vector_type(8))) float v8f;
typedef int v4i __attribute__((vector_size(16)));   // matches builtin's b128 payload type

#define T_DIM 64
#define NB    512
#define NS    32
#define NC    16
#define NSC   48

#define AS1 __attribute__((address_space(1)))
#define AS3 __attribute__((address_space(3)))

#if defined(__has_builtin)
#if __has_builtin(__builtin_amdgcn_global_load_async_to_lds_b128)
#define HAVE_ASYNC_COPY 1
#endif
#if __has_builtin(__builtin_amdgcn_s_wait_asynccnt)
#define HAVE_WAIT_ASYNC_BUILTIN 1
#endif
#endif
#ifndef HAVE_ASYNC_COPY
#define HAVE_ASYNC_COPY 0
#endif
#ifndef HAVE_WAIT_ASYNC_BUILTIN
#define HAVE_WAIT_ASYNC_BUILTIN 0
#endif

// ---------- CDNA5 async global->LDS helpers (ASYNCcnt-tracked DMA path) ----------
__device__ __forceinline__ void async_load_b128(const float* gsrc, float* ldst) {
#if HAVE_ASYNC_COPY
  __builtin_amdgcn_global_load_async_to_lds_b128(
      (AS1 v4i*)(unsigned long long)gsrc,
      (AS3 v4i*)(unsigned int)(unsigned long long)ldst,
      0, 0);
#endif
}
__device__ __forceinline__ void wait_async0() {
#if HAVE_ASYNC_COPY
#if HAVE_WAIT_ASYNC_BUILTIN
  __builtin_amdgcn_s_wait_asynccnt(0);
#else
  asm volatile("s_wait_asynccnt 0x0" ::: "memory");
#endif
#endif
}
__device__ __forceinline__ void stage_Q_async(const float* gQ, float* sQ, int tid) {
#if HAVE_ASYNC_COPY
  for (int i = tid; i < (NSC * NSC) / 4; i += 128)
    async_load_b128(gQ + i * 4, sQ + i * 4);
#endif
}
__device__ __forceinline__ void stage_Q_sync(const float* gQ, float* sQ, int tid) {
  const float4* q4 = (const float4*)gQ;
  float4* s4 = (float4*)sQ;
  for (int i = tid; i < (NSC * NSC) / 4; i += 128) s4[i] = q4[i];
}

// ---------- fp32 WMMA helpers (V_WMMA_F32_16X16X4_F32) ----------
// A-matrix 16x4 layout: lanes 0-15 hold {K=k0, K=k0+1}, lanes 16-31 hold {K=k0+2, K=k0+3}
// B-matrix 4x16 layout: mirrored (K along VGPR/lane-half, N along lanes 0-15)
// C/D 16x16: VGPR r -> row r (lanes 0-15), row r+8 (lanes 16-31)
__device__ __forceinline__ void wmma_acc_f32(
    const float* A, int lda, int arow, int acol,
    const float* B, int ldb, int brow, int bcol,
    int kdim, v8f& acc, int lane)
{
  const int l = lane & 15;
  const int half2 = (lane >> 4) * 2;
#pragma unroll 8
  for (int kk = 0; kk < kdim; kk += 4) {
    v2f a, bb;
    const int ka = acol + kk + half2;
    a.x = A[(arow + l) * lda + ka];
    a.y = A[(arow + l) * lda + ka + 1];
    const int kb = brow + kk + half2;
    bb.x = B[kb * ldb + bcol + l];
    bb.y = B[(kb + 1) * ldb + bcol + l];
    acc = __builtin_amdgcn_wmma_f32_16x16x4_f32(false, a, false, bb,
                                                (short)0, acc, false, false);
  }
}

__device__ __forceinline__ v8f load_tile(const float* C, int ldc, int row, int col, int lane) {
  v8f c;
  const int l = lane & 15;
  const int h = (lane >> 4) * 8;
#pragma unroll
  for (int r = 0; r < 8; ++r) c[r] = C[(row + h + r) * ldc + col + l];
  return c;
}

__device__ __forceinline__ void store_tile(float* D, int ldd, int row, int col, int lane, v8f acc) {
  const int l = lane & 15;
  const int h = (lane >> 4) * 8;
#pragma unroll
  for (int r = 0; r < 8; ++r) D[(row + h + r) * ldd + col + l] = acc[r];
}

// One workgroup per batch: backward Riccati (T sequential steps) then forward rollout.
__global__ __launch_bounds__(128) void lqr_dp_kernel(
    const float* __restrict__ x_init,  // (NB,32)
    const float* __restrict__ Qg,      // (T,NB,48,48)
    const float* __restrict__ pg,      // (T,NB,48)
    const float* __restrict__ Ag,      // (NB,32,32)
    const float* __restrict__ Bg,      // (NB,32,16)
    const float* __restrict__ c1g,     // (NB,32)
    float* __restrict__ xs,            // (T,NB,32)
    float* __restrict__ us,            // (T,NB,16)
    float* __restrict__ objs,          // (T,NB)
    float* __restrict__ tau,           // (T,NB,48)
    float* __restrict__ Kws,           // (T,NB,16,32) scratch
    float* __restrict__ kws)           // (T,NB,16)   scratch
{
  __shared__ float sF [NS * NSC];   // F = [A|B], 32x48
  __shared__ float sFt[NSC * NS];   // F^T, 48x32
  __shared__ float sV [NS * NS];    // value matrix
  __shared__ float sVn[NS * NS];
  __shared__ float sM [NS * NSC];   // V @ F
  __shared__ __align__(16) float sQbuf[2][NSC * NSC];  // double-buffered Q[t,b] / Qt
  __shared__ float sT1[NC * NS];    // Quu@K, then Qux + Quu@K
  __shared__ float sK [NC * NS];
  __shared__ float sKt[NS * NC];
  __shared__ float sAug[NC * 32];   // [Quu | I] Gauss-Jordan
  __shared__ float smul[NC];
  __shared__ float sc1[NS];
  __shared__ float sv [NS], svn[NS], svf[NS];
  __shared__ float sqt[NSC];
  __shared__ float sk [NC], sw[NC];
  __shared__ float sxu[NSC], sxn[NS], sred[NSC];
  __shared__ float spiv;

  const int b    = blockIdx.x;
  const int tid  = threadIdx.x;
  const int lane = tid & 31;
  const int wave = tid >> 5;

  // ---- init F, Ft, c1, V=0, v=0 ----
  for (int i = tid; i < NS * NS; i += 128) {
    sF[(i >> 5) * NSC + (i & 31)] = Ag[(size_t)b * NS * NS + i];
    sV[i] = 0.f;
  }
  for (int i = tid; i < NS * NC; i += 128)
    sF[(i >> 4) * NSC + NS + (i & 15)] = Bg[(size_t)b * NS * NC + i];
  if (tid < NS) { sc1[tid] = c1g[b * NS + tid]; sv[tid] = 0.f; }
  __syncthreads();
  for (int i = tid; i < NSC * NS; i += 128) {
    const int r = i >> 5, c = i & 31;
    sFt[r * NS + c] = sF[c * NSC + r];
  }
  __syncthreads();

#if HAVE_ASYNC_COPY
  // prologue: async-stage Q[T-1,b] into its parity buffer
  stage_Q_async(Qg + ((size_t)(T_DIM - 1) * NB + b) * (NSC * NSC),
                sQbuf[(T_DIM - 1) & 1], tid);
#endif

  // ================= backward Riccati recursion =================
  for (int t = T_DIM - 1; t >= 0; --t) {
    float* sQ = sQbuf[t & 1];
    const size_t qbase = ((size_t)t * NB + b) * (NSC * NSC);

#if HAVE_ASYNC_COPY
    wait_async0();          // per-wave: my Q[t] transfers have landed in LDS
    __syncthreads();        // all waves' transfers visible
    if (t > 0)              // overlap Q[t-1] DMA with this whole step's compute
      stage_Q_async(Qg + qbase - (size_t)NB * NSC * NSC, sQbuf[(t - 1) & 1], tid);
#else
    stage_Q_sync(Qg + qbase, sQ, tid);
#endif
    if (t > 1) {            // keep L2 warm two steps ahead
      const float* nq = Qg + qbase - 2 * (size_t)NB * NSC * NSC;
      if (tid < 72) __builtin_prefetch(nq + tid * 32, 0, 1);
    }
    __syncthreads();

    // M = V @ F  (32x48, 2x3 tiles, K=32)
    for (int tile = wave; tile < 6; tile += 4) {
      const int ti = tile / 3, tj = tile % 3;
      v8f acc = {};
      wmma_acc_f32(sV, NS, ti * 16, 0, sF, NSC, 0, tj * 16, NS, acc, lane);
      store_tile(sM, NSC, ti * 16, tj * 16, lane, acc);
    }
    // vf = V@c1 + v
    if (tid < NS) {
      float s = sv[tid];
      for (int j = 0; j < NS; ++j) s += sV[tid * NS + j] * sc1[j];
      svf[tid] = s;
    }
    __syncthreads();

    // Qt = Q + F^T @ M  (48x48, 3x3 tiles, K=32)
    for (int tile = wave; tile < 9; tile += 4) {
      const int ti = tile / 3, tj = tile % 3;
      v8f acc = load_tile(sQ, NSC, ti * 16, tj * 16, lane);
      wmma_acc_f32(sFt, NS, ti * 16, 0, sM, NSC, 0, tj * 16, NS, acc, lane);
      store_tile(sQ, NSC, ti * 16, tj * 16, lane, acc);
    }
    // qt = p + F^T @ vf
    if (tid < NSC) {
      float s = pg[((size_t)t * NB + b) * NSC + tid];
      for (int j = 0; j < NS; ++j) s += sFt[tid * NS + j] * svf[j];
      sqt[tid] = s;
    }
    __syncthreads();

    // ---- Quu^{-1} (16x16 SPD) via Gauss-Jordan on [Quu | I] ----
    for (int i = tid; i < NC * 32; i += 128) {
      const int r = i >> 5, c = i & 31;
      sAug[i] = (c < NC) ? sQ[(NS + r) * NSC + NS + c] : (((c - NC) == r) ? 1.f : 0.f);
    }
    __syncthreads();
    for (int piv = 0; piv < NC; ++piv) {
      if (tid == 0) spiv = 1.f / sAug[piv * 32 + piv];
      __syncthreads();
      if (tid < 32) sAug[piv * 32 + tid] *= spiv;
      else if (tid >= 32 && tid < 32 + NC && (tid - 32) != piv)
        smul[tid - 32] = sAug[(tid - 32) * 32 + piv];
      __syncthreads();
      for (int i = tid; i < NC * 32; i += 128) {
        const int r = i >> 5;
        if (r != piv) sAug[i] -= smul[r] * sAug[piv * 32 + (i & 31)];
      }
      __syncthreads();
    }

    // K = -(Quu_inv @ Qux)  (16x32, K=16); also K^T
    for (int tj = wave; tj < 2; tj += 4) {
      v8f acc = {};
      wmma_acc_f32(sAug, 32, 0, NC, sQ, NSC, NS, tj * 16, NC, acc, lane);
      const int l = lane & 15, h = (lane >> 4) * 8;
#pragma unroll
      for (int r = 0; r < 8; ++r) {
        const float val = -acc[r];
        const int row = h + r, col = tj * 16 + l;
        sK[row * NS + col] = val;
        sKt[col * NC + row] = val;
      }
    }
    // k = -(Quu_inv @ qu)
    if (tid < NC) {
      float s = 0.f;
      for (int j = 0; j < NC; ++j) s += sAug[tid * 32 + NC + j] * sqt[NS + j];
      sk[tid] = -s;
    }
    __syncthreads();

    { // spill gains for forward pass
      const size_t kb = (size_t)t * NB + b;
      for (int i = tid; i < NC * NS; i += 128) Kws[kb * (NC * NS) + i] = sK[i];
      if (tid < NC) kws[kb * NC + tid] = sk[tid];
    }

    // T1 = Quu @ K  (16x32, K=16)
    for (int tj = wave; tj < 2; tj += 4) {
      v8f acc = {};
      wmma_acc_f32(sQ, NSC, NS, NS, sK, NS, 0, tj * 16, NC, acc, lane);
      store_tile(sT1, NS, 0, tj * 16, lane, acc);
    }
    // w = qu + Quu @ k
    if (tid < NC) {
      float s = sqt[NS + tid];
      for (int j = 0; j < NC; ++j) s += sQ[(NS + tid) * NSC + NS + j] * sk[j];
      sw[tid] = s;
    }
    __syncthreads();
    // T2 = Qux + T1   (in place)
    for (int i = tid; i < NC * NS; i += 128)
      sT1[i] += sQ[(NS + (i >> 5)) * NSC + (i & 31)];
    __syncthreads();

    // Vn = Qxx + Qxu@K + K^T@(Qux + Quu@K)   (32x32, 2x2 tiles, K=16 each)
    for (int tile = wave; tile < 4; tile += 4) {
      const int ti = tile >> 1, tj = tile & 1;
      v8f acc = load_tile(sQ, NSC, ti * 16, tj * 16, lane);
      wmma_acc_f32(sQ, NSC, ti * 16, NS, sK, NS, 0, tj * 16, NC, acc, lane);
      wmma_acc_f32(sKt, NC, ti * 16, 0, sT1, NS, 0, tj * 16, NC, acc, lane);
      store_tile(sVn, NS, ti * 16, tj * 16, lane, acc);
    }
    // vn = qx + Qxu@k + K^T@w
    if (tid < NS) {
      float s = sqt[tid];
      for (int j = 0; j < NC; ++j) s += sQ[tid * NSC + NS + j] * sk[j];
      for (int j = 0; j < NC; ++j) s += sKt[tid * NC + j] * sw[j];
      svn[tid] = s;
    }
    __syncthreads();
    for (int i = tid; i < NS * NS; i += 128) sV[i] = sVn[i];
    if (tid < NS) sv[tid] = svn[tid];
    __syncthreads();
  }

  // ================= forward rollout =================
  if (tid < NS) sxu[tid] = x_init[b * NS + tid];
  __syncthreads();
  for (int t = 0; t < T_DIM; ++t) {
    const size_t kb = (size_t)t * NB + b;
    // u = K x + k
    if (tid < NC) {
      float s = kws[kb * NC + tid];
      const float* Krow = Kws + kb * (NC * NS) + tid * NS;
      for (int j = 0; j < NS; ++j) s += Krow[j] * sxu[j];
      sxu[NS + tid] = s;
    }
    __syncthreads();
    // obj partials: xu_i * (0.5*(Q xu)_i + p_i)
    if (tid < NSC) {
      const float* Qrow = Qg + kb * (NSC * NSC) + tid * NSC;
      float qd = 0.f;
      for (int j = 0; j < NSC; ++j) qd += Qrow[j] * sxu[j];
      sred[tid] = sxu[tid] * (0.5f * qd + pg[kb * NSC + tid]);
    }
    // x_next = F @ xu + c1  (done by a disjoint thread group)
    if (tid >= 64 && tid < 96) {
      const int i = tid - 64;
      float s = sc1[i];
      for (int j = 0; j < NSC; ++j) s += sF[i * NSC + j] * sxu[j];
      sxn[i] = s;
    }
    __syncthreads();
    if (tid < NS)  xs[kb * NS + tid]  = sxu[tid];
    if (tid < NC)  us[kb * NC + tid]  = sxu[NS + tid];
    if (tid < NSC) tau[kb * NSC + tid] = sxu[tid];
    if (tid == 0) {
      float o = 0.f;
      for (int j = 0; j < NSC; ++j) o += sred[j];
      objs[kb] = o;
    }
    __syncthreads();
    if (tid < NS) sxu[tid] = sxn[tid];
    __syncthreads();
  }
}

extern "C" void kernel_launch(void* const* d_in, const int* in_sizes, int n_in,
                              void* d_out, int out_size, void* d_ws, size_t ws_size,
                              hipStream_t stream) {
  const float* x_init = (const float*)d_in[0];
  const float* Q      = (const float*)d_in[1];
  const float* p      = (const float*)d_in[2];
  const float* A      = (const float*)d_in[3];
  const float* B      = (const float*)d_in[4];
  const float* c1     = (const float*)d_in[5];

  float* out  = (float*)d_out;
  float* xs   = out;
  float* us   = xs   + (size_t)T_DIM * NB * NS;
  float* objs = us   + (size_t)T_DIM * NB * NC;
  float* tau  = objs + (size_t)T_DIM * NB;

  float* Kws = (float*)d_ws;                          // T*NB*16*32 floats
  float* kws = Kws + (size_t)T_DIM * NB * NC * NS;    // T*NB*16 floats

  lqr_dp_kernel<<<NB, 128, 0, stream>>>(x_init, Q, p, A, B, c1,
                                        xs, us, objs, tau, Kws, kws);
}